// HivemindAgent_65730179498605
// MI455X (gfx1250) — compile-verified
//
#include <hip/hip_runtime.h>
#include <hip/hip_bf16.h>

// ---------------------------------------------------------------------------
// Key observation: in the reference, cross-attention has a single KV token, so
// softmax over the size-1 axis is exactly 1.0 and the attention output equals
// v broadcast over all query positions. Therefore pred/act depend ONLY on:
//   ctx  = context_batch @ ctx_w + ctx_b
//   v    = ctx  @ ca_wv + ca_bv
//   last = v    @ ca_wo + ca_bo
//   pred = last @ pred_w[spec] + pred_b[spec]
//   act  = last @ act_w[spec]  + act_b[spec]
// The embedding + 4-layer transformer is dead code w.r.t. the outputs.
// One workgroup (32 waves), one reused 32KB LDS buffer, f32 WMMA for the two
// 16x512x512 GEMMs (M=16 == one WMMA tile row; exact f32 numerics).
// ---------------------------------------------------------------------------

typedef __attribute__((ext_vector_type(2))) float v2f;
typedef __attribute__((ext_vector_type(8))) float v8f;

#define BSZ  16
#define DIM  512
#define OUTD 64

__launch_bounds__(1024, 1)
__global__ void hivemind_tail_kernel(const float* __restrict__ context_batch, // [16,2]
                                     const int*   __restrict__ specialist_ids,// [16]
                                     const float* __restrict__ ctx_w,  // [2,512]
                                     const float* __restrict__ ctx_b,  // [512]
                                     const float* __restrict__ ca_wv,  // [512,512]
                                     const float* __restrict__ ca_bv,  // [512]
                                     const float* __restrict__ ca_wo,  // [512,512]
                                     const float* __restrict__ ca_bo,  // [512]
                                     const float* __restrict__ pred_w, // [4,512,64]
                                     const float* __restrict__ pred_b, // [4,64]
                                     const float* __restrict__ act_w,  // [4,512,64]
                                     const float* __restrict__ act_b,  // [4,64]
                                     float*       __restrict__ out)    // [2048] = pred(1024) ++ act(1024)
{
    __shared__ float buf[BSZ][DIM]; // 32 KB; holds ctx, then v, then last

    const int tid  = threadIdx.x;       // 0..1023
    const int lane = tid & 31;
    const int wave = tid >> 5;          // 0..31  -> N-tile (16 cols each) of 512

    // ---- stage 0: ctx = context_batch @ ctx_w + ctx_b  (K = 2, pure VALU) --
    for (int i = tid; i < BSZ * DIM; i += 1024) {
        const int b = i >> 9;
        const int d = i & (DIM - 1);
        buf[b][d] = fmaf(context_batch[2 * b + 0], ctx_w[d],
                    fmaf(context_batch[2 * b + 1], ctx_w[DIM + d], ctx_b[d]));
    }
    // Pre-warm second GEMM's weights into cache while GEMM 1 runs.
    __builtin_prefetch(ca_wo + (size_t)tid * 256, 0, 1);
    __syncthreads();

    // Fragment geometry for V_WMMA_F32_16X16X4_F32 (ISA 7.12.2):
    //   A 16x4 f32 : lane<16 -> row=lane,    {K=k+0, K=k+1}
    //                lane>=16 -> row=lane-16, {K=k+2, K=k+3}
    //   B 4x16 f32 : symmetric by column (col = lane & 15)
    //   C/D 16x16  : lane&15 = N, vgpr r -> M = r + 8*(lane>=16)
    const int row    = lane & 15;            // A row / D column-within-tile
    const int khalf  = (lane >> 4) << 1;     // 0 or 2
    const int mbase  = (lane >> 4) << 3;     // 0 or 8 for C/D rows
    const int col    = (wave << 4) + row;    // global N column for B frag / bias / D

    // ---- stage 1: v = ctx @ ca_wv + ca_bv  (M=16, N=512, K=512; WMMA) -----
    v8f acc = {};
    for (int kk = 0; kk < DIM; kk += 4) {
        const int k0 = kk + khalf;
        v2f a; a.x = buf[row][k0]; a.y = buf[row][k0 + 1];
        v2f bfr; bfr.x = ca_wv[(size_t)k0 * DIM + col];
                 bfr.y = ca_wv[(size_t)(k0 + 1) * DIM + col];
        acc = __builtin_amdgcn_wmma_f32_16x16x4_f32(
            /*neg_a=*/false, a, /*neg_b=*/false, bfr,
            /*c_mod=*/(short)0, acc, /*reuse_a=*/false, /*reuse_b=*/false);
    }
    {
        const float bias = ca_bv[col];
        __syncthreads();                      // all waves done reading ctx
        #pragma unroll
        for (int r = 0; r < 8; ++r)
            buf[mbase + r][col] = acc[r] + bias;   // buf := v
        __syncthreads();
    }

    // ---- stage 2: last = v @ ca_wo + ca_bo  (same shape, WMMA) ------------
    v8f acc2 = {};
    for (int kk = 0; kk < DIM; kk += 4) {
        const int k0 = kk + khalf;
        v2f a; a.x = buf[row][k0]; a.y = buf[row][k0 + 1];
        v2f bfr; bfr.x = ca_wo[(size_t)k0 * DIM + col];
                 bfr.y = ca_wo[(size_t)(k0 + 1) * DIM + col];
        acc2 = __builtin_amdgcn_wmma_f32_16x16x4_f32(
            false, a, false, bfr, (short)0, acc2, false, false);
    }
    {
        const float bias = ca_bo[col];
        __syncthreads();                      // all waves done reading v
        #pragma unroll
        for (int r = 0; r < 8; ++r)
            buf[mbase + r][col] = acc2[r] + bias;  // buf := last
        __syncthreads();
    }

    // ---- stage 3: specialist heads (batched vec-mat, gathered weights) ----
    // task t in [0,1024): pred, b = t/64, o = t%64 ; rep 1: act.
    {
        const int b = tid >> 6;               // 0..15
        const int o = tid & 63;               // 0..63
        const int spec = specialist_ids[b];
        #pragma unroll
        for (int rep = 0; rep < 2; ++rep) {
            const float* __restrict__ W  = (rep == 0 ? pred_w : act_w) + (size_t)spec * DIM * OUTD;
            const float* __restrict__ Bb = (rep == 0 ? pred_b : act_b) + spec * OUTD;
            float s = Bb[o];
            #pragma unroll 8
            for (int d = 0; d < DIM; ++d)
                s = fmaf(buf[b][d], W[(size_t)d * OUTD + o], s);  // coalesced over o
            out[rep * 1024 + tid] = s;
        }
    }
}

extern "C" void kernel_launch(void* const* d_in, const int* in_sizes, int n_in,
                              void* d_out, int out_size, void* d_ws, size_t ws_size,
                              hipStream_t stream) {
    (void)in_sizes; (void)n_in; (void)out_size; (void)d_ws; (void)ws_size;
    // setup_inputs order:
    // 0:x_batch 1:context_batch 2:specialist_ids 3:emb_table 4:wqkv 5:bqkv
    // 6:wo 7:bo 8:ln1_g 9:ln1_b 10:w1 11:b1 12:w2 13:b2 14:ln2_g 15:ln2_b
    // 16:ctx_w 17:ctx_b 18:ca_wq 19:ca_bq 20:ca_wk 21:ca_bk 22:ca_wv 23:ca_bv
    // 24:ca_wo 25:ca_bo 26:pred_w 27:pred_b 28:act_w 29:act_b
    const float* context_batch = (const float*)d_in[1];
    const int*   spec          = (const int*)  d_in[2];
    const float* ctx_w  = (const float*)d_in[16];
    const float* ctx_b  = (const float*)d_in[17];
    const float* ca_wv  = (const float*)d_in[22];
    const float* ca_bv  = (const float*)d_in[23];
    const float* ca_wo  = (const float*)d_in[24];
    const float* ca_bo  = (const float*)d_in[25];
    const float* pred_w = (const float*)d_in[26];
    const float* pred_b = (const float*)d_in[27];
    const float* act_w  = (const float*)d_in[28];
    const float* act_b  = (const float*)d_in[29];

    hivemind_tail_kernel<<<1, 1024, 0, stream>>>(
        context_batch, spec, ctx_w, ctx_b, ca_wv, ca_bv, ca_wo, ca_bo,
        pred_w, pred_b, act_w, act_b, (float*)d_out);
}